// Sampler_65609920414297
// MI455X (gfx1250) — compile-verified
//
#include <hip/hip_runtime.h>
#include <stdint.h>

// Problem constants (fixed by the reference): B=256, V=128000, L=2048
#define BLOCK      1024                 // 32 wave32s per workgroup (one row each)
#define VOCAB      128000
#define HIST_L     2048
#define N4         (VOCAB / 4)          // 32000 float4 per row
#define FULL_K     (N4 / BLOCK)         // 31 full strided iterations
#define MASK_WORDS (VOCAB / 32)         // 4000 words = 16 KB LDS bitmask
#define NWAVES     (BLOCK / 32)         // 32 waves

typedef float f4 __attribute__((ext_vector_type(4)));

// Tie-aware merge: adopt (ov,oi) if value greater, or equal with smaller index
// (reproduces jnp.argmax first-occurrence semantics across lanes/accumulators).
__device__ __forceinline__ void merge(float& v, int& i, float ov, int oi) {
    if (ov > v || (ov == v && oi < i)) { v = ov; i = oi; }
}

__global__ __launch_bounds__(BLOCK)
void sampler_row_kernel(const float* __restrict__ logits,
                        const int*   __restrict__ token_ids,
                        const float* __restrict__ pres_pen,
                        const float* __restrict__ temps,
                        const float* __restrict__ gumbel,
                        int* __restrict__ out)
{
    __shared__ uint32_t s_mask[MASK_WORDS];   // presence bitmask, 16 KB
    __shared__ int      s_tok[HIST_L];        // staged token history, 8 KB
    __shared__ float    s_wv[NWAVES];
    __shared__ int      s_wi[NWAVES];

    const int row = blockIdx.x;
    const int tid = threadIdx.x;

    // 1) zero the presence bitmask
    for (int i = tid; i < MASK_WORDS; i += BLOCK) s_mask[i] = 0u;

    // 2) stage token history global->LDS via CDNA5 async-DMA path.
    //    Each wave waits only on its own ASYNCcnt; the workgroup barrier
    //    below publishes all waves' data before use.
    {
        const int* g = token_ids + (size_t)row * HIST_L;
        for (int j = tid; j < HIST_L; j += BLOCK) {
            uint32_t laddr = (uint32_t)(uintptr_t)(&s_tok[j]);
            uint64_t gaddr = (uint64_t)(uintptr_t)(g + j);
            asm volatile("global_load_async_to_lds_b32 %0, %1, off"
                         :: "v"(laddr), "v"(gaddr) : "memory");
        }
        asm volatile("s_wait_asynccnt 0" ::: "memory");
    }
    __syncthreads();

    // 3) scatter presence bits (ds_or_b32 atomics in LDS)
    for (int j = tid; j < HIST_L; j += BLOCK) {
        int t = s_tok[j];
        atomicOr(&s_mask[t >> 5], 1u << (t & 31));
    }
    __syncthreads();

    // Per-row uniform scalars (wave-uniform -> scalar loads)
    const float pp     = pres_pen[row];
    const float temp   = temps[row];
    const bool  greedy = temp < 1e-5f;
    const float inv_t  = 1.0f / (greedy ? 1.0f : temp);

    const f4* lg = (const f4*)(logits + (size_t)row * VOCAB);
    const f4* gb = (const f4*)(gumbel + (size_t)row * VOCAB);

    // 4 independent accumulators (one per f4 lane) -> 4x shorter select chain.
    // Each visits strictly increasing indices, so strict '>' keeps the first
    // occurrence, matching jnp.argmax.
    float bv0 = -__builtin_inff(), bv1 = -__builtin_inff(),
          bv2 = -__builtin_inff(), bv3 = -__builtin_inff();
    int   bi0 = 0, bi1 = 1, bi2 = 2, bi3 = 3;

    if (greedy) {
        // argmax(penalized) — skip the gumbel stream entirely (saves 512 KB of reads per row)
        auto body = [&](int i) {
            f4 lv = __builtin_nontemporal_load(&lg[i]);      // TH=NT streaming
            uint32_t bits = s_mask[i >> 3] >> ((i & 7) * 4); // 4 presence bits
            int e = i << 2;
            float v0 = lv.x - ((bits & 1u) ? pp : 0.0f);
            float v1 = lv.y - ((bits & 2u) ? pp : 0.0f);
            float v2 = lv.z - ((bits & 4u) ? pp : 0.0f);
            float v3 = lv.w - ((bits & 8u) ? pp : 0.0f);
            if (v0 > bv0) { bv0 = v0; bi0 = e + 0; }
            if (v1 > bv1) { bv1 = v1; bi1 = e + 1; }
            if (v2 > bv2) { bv2 = v2; bi2 = e + 2; }
            if (v3 > bv3) { bv3 = v3; bi3 = e + 3; }
        };
        int i = tid;
        #pragma unroll 4
        for (int k = 0; k < FULL_K; ++k, i += BLOCK) body(i);
        if (i < N4) body(i);
    } else {
        // argmax(penalized/t + gumbel); 1/t hoisted per row (monotone scale)
        auto body = [&](int i) {
            f4 lv = __builtin_nontemporal_load(&lg[i]);
            f4 gv = __builtin_nontemporal_load(&gb[i]);
            uint32_t bits = s_mask[i >> 3] >> ((i & 7) * 4);
            int e = i << 2;
            float v0 = fmaf(lv.x - ((bits & 1u) ? pp : 0.0f), inv_t, gv.x);
            float v1 = fmaf(lv.y - ((bits & 2u) ? pp : 0.0f), inv_t, gv.y);
            float v2 = fmaf(lv.z - ((bits & 4u) ? pp : 0.0f), inv_t, gv.z);
            float v3 = fmaf(lv.w - ((bits & 8u) ? pp : 0.0f), inv_t, gv.w);
            if (v0 > bv0) { bv0 = v0; bi0 = e + 0; }
            if (v1 > bv1) { bv1 = v1; bi1 = e + 1; }
            if (v2 > bv2) { bv2 = v2; bi2 = e + 2; }
            if (v3 > bv3) { bv3 = v3; bi3 = e + 3; }
        };
        int i = tid;
        #pragma unroll 4
        for (int k = 0; k < FULL_K; ++k, i += BLOCK) body(i);
        if (i < N4) body(i);
    }

    // 4) merge the 4 accumulators (tie -> smaller index)
    float bv = bv0; int bi = bi0;
    merge(bv, bi, bv1, bi1);
    merge(bv, bi, bv2, bi2);
    merge(bv, bi, bv3, bi3);

    // 5) wave32 reduction
    #pragma unroll
    for (int off = 16; off > 0; off >>= 1) {
        float ov = __shfl_xor(bv, off, 32);
        int   oi = __shfl_xor(bi, off, 32);
        merge(bv, bi, ov, oi);
    }
    const int wave = tid >> 5;
    const int lane = tid & 31;
    if (lane == 0) { s_wv[wave] = bv; s_wi[wave] = bi; }
    __syncthreads();

    // 6) cross-wave reduction by wave 0 (exactly 32 partials)
    if (wave == 0) {
        bv = s_wv[lane];
        bi = s_wi[lane];
        #pragma unroll
        for (int off = 16; off > 0; off >>= 1) {
            float ov = __shfl_xor(bv, off, 32);
            int   oi = __shfl_xor(bi, off, 32);
            merge(bv, bi, ov, oi);
        }
        if (lane == 0) out[row] = bi;
    }
}

extern "C" void kernel_launch(void* const* d_in, const int* in_sizes, int n_in,
                              void* d_out, int out_size, void* d_ws, size_t ws_size,
                              hipStream_t stream) {
    (void)n_in; (void)out_size; (void)d_ws; (void)ws_size;
    const float* logits    = (const float*)d_in[0];   // [B, V] f32
    const int*   token_ids = (const int*)  d_in[1];   // [B, L] i32
    const float* pres_pen  = (const float*)d_in[2];   // [B]    f32
    const float* temps     = (const float*)d_in[3];   // [B]    f32
    const float* gumbel    = (const float*)d_in[4];   // [B, V] f32
    int*         out       = (int*)d_out;             // [B]    i32

    const int B = in_sizes[2];                        // 256 rows
    sampler_row_kernel<<<B, BLOCK, 0, stream>>>(logits, token_ids, pres_pen,
                                                temps, gumbel, out);
}